// InteractGraph_9466107920427
// MI455X (gfx1250) — compile-verified
//
#include <hip/hip_runtime.h>
#include <math.h>

// ---------------- problem constants ----------------
#define B_    32
#define N_    64
#define NH_   16
#define R_    1024
#define REP_  1024
#define M_    1008     // 16*63
#define NCLS_ 117

typedef __attribute__((ext_vector_type(2))) float v2f;
typedef __attribute__((ext_vector_type(8))) float v8f;

__device__ __forceinline__ v8f wmma_f32_4(v2f a, v2f b, v8f c) {
  // V_WMMA_F32_16X16X4_F32 : D = A(16x4,f32) * B(4x16,f32) + C(16x16,f32)
  return __builtin_amdgcn_wmma_f32_16x16x4_f32(false, a, false, b, (short)0, c,
                                               false, false);
}

__device__ __forceinline__ void pair_xy(int p, int& x, int& y) {
  x = p / 63;
  int r = p % 63;
  y = r + (r >= x ? 1 : 0);
}

// ---------------- generic batched WMMA GEMM ----------------
// C[b] = act( A[b] (rows x K, row-major lda) @ W[b] + bias )
// W element (k,n) at  W + n*nstride + k*kstride.
//   transposed weights: kstride==1, nstride==K  -> contiguous v2f B-loads
//   row-major weights : kstride==ldw, nstride==1
// one wave per 16x16 output tile; fp32 WMMA, K stepped by 4.
__global__ void __launch_bounds__(32)
wmma_gemm_kernel(const float* __restrict__ A, const float* __restrict__ W,
                 const float* __restrict__ bias, float* __restrict__ C,
                 int K, int lda, int kstride, int nstride, int ldc,
                 long long sA, long long sW, long long sC, int relu) {
  const int lane  = threadIdx.x;           // 0..31
  const int ntile = blockIdx.x;
  const int mtile = blockIdx.y;
  const int b     = blockIdx.z;

  const int mlo  = lane & 15;              // A row within tile / B col within tile
  const int khi  = (lane >> 4) * 2;        // K sub-offset per half-wave
  const int ncol = ntile * 16 + mlo;

  const float* ap = A + (long long)b * sA +
                    (long long)(mtile * 16 + mlo) * lda + khi;
  const float* wp = W + (long long)b * sW + (long long)ncol * nstride +
                    (long long)khi * kstride;

  v8f acc = {0.f, 0.f, 0.f, 0.f, 0.f, 0.f, 0.f, 0.f};
  if (kstride == 1) {
    // transposed weights: single 8B load per fragment
    for (int k = 0; k < K; k += 4, ap += 4, wp += 4)
      acc = wmma_f32_4(*(const v2f*)ap, *(const v2f*)wp, acc);
  } else {
    const long long wstep = 4ll * kstride;
    for (int k = 0; k < K; k += 4, ap += 4, wp += wstep) {
      v2f bv;
      bv.x = wp[0];
      bv.y = wp[kstride];
      acc = wmma_f32_4(*(const v2f*)ap, bv, acc);
    }
  }

  const float bb = bias ? bias[ncol] : 0.0f;
  float* Cp = C + (long long)b * sC + ncol;
  const int mbase = mtile * 16 + (lane >> 4) * 8;    // D elem i -> row i + 8*(lane/16)
#pragma unroll
  for (int i = 0; i < 8; ++i) {
    float v = acc[i] + bb;
    if (relu) v = fmaxf(v, 0.0f);
    Cp[(long long)(mbase + i) * ldc] = v;
  }
}

// ---------------- fused pair MLP ----------------
// one block (4 waves) per 16 pairs:
//  h = relu(U[x] + V[y] + b1)  (LDS, 16x1024 fp32)
//  h2 = relu(h @ W2 + b2)      (WMMA over 512 cols, 128 cols per wave, W2^T reads)
//  w  = h2 @ W3 + b3 ; A[x,y] = sigmoid(w)  (fused reduction)
#define HPAD 1032  // row pitch (even -> 8B-aligned float2 loads, staggers banks)
__global__ void __launch_bounds__(128)
pair_mlp_kernel(const float* __restrict__ U, const float* __restrict__ V,
                const float* __restrict__ b1, const float* __restrict__ W2T,
                const float* __restrict__ b2, const float* __restrict__ W3,
                const float* __restrict__ b3, float* __restrict__ Abuf,
                float* __restrict__ AT) {
  extern __shared__ float sm[];
  float* h    = sm;                 // [16][HPAD]
  float* wsum = sm + 16 * HPAD;     // [16]

  const int tid  = threadIdx.x;
  const int tile = blockIdx.x;      // 0..62
  const int b    = blockIdx.y;

  // build h tile: 16 rows x 1024
  for (int i = tid; i < 16 * 1024; i += 128) {
    const int row = i >> 10;
    const int c   = i & 1023;
    int x, y;
    pair_xy(tile * 16 + row, x, y);
    float v = U[((long long)b * N_ + x) * R_ + c] +
              V[((long long)b * N_ + y) * R_ + c] + b1[c];
    h[row * HPAD + c] = fmaxf(v, 0.0f);
  }
  if (tid < 16) wsum[tid] = 0.0f;
  __syncthreads();

  const int lane = tid & 31;
  const int wv   = tid >> 5;        // wave 0..3 -> 128 cols each
  const int mlo  = lane & 15;
  const int khi  = (lane >> 4) * 2;

  float part[8] = {0.f, 0.f, 0.f, 0.f, 0.f, 0.f, 0.f, 0.f};
#pragma unroll 1
  for (int t = 0; t < 8; ++t) {
    const int ncol = wv * 128 + t * 16 + mlo;
    const float* hp = h + mlo * HPAD + khi;
    const float* wp = W2T + (long long)ncol * REP_ + khi;  // W2^T row = column
    v8f acc = {0.f, 0.f, 0.f, 0.f, 0.f, 0.f, 0.f, 0.f};
    for (int k = 0; k < REP_; k += 4, hp += 4, wp += 4)
      acc = wmma_f32_4(*(const v2f*)hp, *(const v2f*)wp, acc);
    const float w3 = W3[ncol];
    const float bb = b2[ncol];
#pragma unroll
    for (int i = 0; i < 8; ++i) part[i] += fmaxf(acc[i] + bb, 0.0f) * w3;
  }
  const int moff = (lane >> 4) * 8;
#pragma unroll
  for (int i = 0; i < 8; ++i) atomicAdd(&wsum[moff + i], part[i]);
  __syncthreads();

  if (tid < 16) {
    const float w = wsum[tid] + b3[0];
    const float a = 1.0f / (1.0f + expf(-w));
    int x, y;
    pair_xy(tile * 16 + tid, x, y);
    Abuf[((long long)b * NH_ + x) * N_ + y] = a;
    AT[((long long)b * N_ + y) * NH_ + x]   = a;
  }
}

// ---------------- utility kernels ----------------
__global__ void zero_kernel(float* p, long long n) {
  for (long long i = blockIdx.x * (long long)blockDim.x + threadIdx.x; i < n;
       i += (long long)gridDim.x * blockDim.x)
    p[i] = 0.0f;
}

// dst[N][K] = src[K][N]^T  (coalesced writes; reads served by L2)
__global__ void transpose_kernel(float* __restrict__ dst,
                                 const float* __restrict__ src, int K, int N) {
  const long long total = (long long)K * N;
  for (long long i = blockIdx.x * (long long)blockDim.x + threadIdx.x; i < total;
       i += (long long)gridDim.x * blockDim.x) {
    const int k = (int)(i % K);
    const int n = (int)(i / K);
    dst[i] = src[(long long)k * N + n];
  }
}

__global__ void copy_rows_kernel(float* __restrict__ dst, const float* __restrict__ src,
                                 int rows, int cols, long long dSB, int dLd,
                                 long long sSB, int sLd, int nb) {
  const long long total = (long long)nb * rows * cols;
  for (long long i = blockIdx.x * (long long)blockDim.x + threadIdx.x; i < total;
       i += (long long)gridDim.x * blockDim.x) {
    const int c = (int)(i % cols);
    const long long t = i / cols;
    const int r = (int)(t % rows);
    const int b = (int)(t / rows);
    dst[b * dSB + (long long)r * dLd + c] = src[b * sSB + (long long)r * sLd + c];
  }
}

__device__ __forceinline__ float lis_f(float x) {
  return 8.3f / (1.0f + expf(12.0f - 10.0f * x));
}

// one block per output pair row: pair gather (float4), coords, prior
__global__ void __launch_bounds__(128)
finalize_kernel(const float* __restrict__ enc, const float* __restrict__ coords,
                const int* __restrict__ labels, const float* __restrict__ scores,
                const float* __restrict__ Abuf, float* __restrict__ out) {
  const int gid = blockIdx.x;         // 0 .. B*M-1
  const int b = gid / M_;
  const int m = gid % M_;
  int x, y;
  pair_xy(m, x, y);
  const int tid = threadIdx.x;

  const float4* ex = (const float4*)(enc + ((long long)b * N_ + x) * R_);
  const float4* ey = (const float4*)(enc + ((long long)b * N_ + y) * R_);
  float4* op = (float4*)(out + (long long)gid * (2 * R_));
#pragma unroll
  for (int j = tid; j < 512; j += 128)   // 2048 floats = 512 float4
    op[j] = (j < 256) ? ex[j] : ey[j - 256];

  const long long bh_off = (long long)B_ * M_ * (2 * R_);
  const long long bo_off = bh_off + (long long)B_ * M_ * 4;
  const long long pr_off = bo_off + (long long)B_ * M_ * 4;
  if (tid < 4) {
    out[bh_off + (long long)gid * 4 + tid] = coords[((long long)b * N_ + x) * 4 + tid];
    out[bo_off + (long long)gid * 4 + tid] = coords[((long long)b * N_ + y) * 4 + tid];
  }
  const int lab = labels[b * N_ + y];
  const float pv = Abuf[((long long)b * NH_ + x) * N_ + y] *
                   lis_f(scores[b * N_ + x]) * lis_f(scores[b * N_ + y]);
  for (int c = tid; c < NCLS_; c += 128)
    out[pr_off + (long long)gid * NCLS_ + c] = (c == lab) ? pv : 0.0f;
}

// ---------------- host orchestration ----------------
static inline void gemm(hipStream_t s, const float* A, const float* W,
                        const float* bias, float* C, int rows, int K, int N,
                        int lda, int kstride, int nstride, int ldc, long long sA,
                        long long sW, long long sC, int relu) {
  dim3 g(N / 16, rows / 16, B_);
  wmma_gemm_kernel<<<g, dim3(32), 0, s>>>(A, W, bias, C, K, lda, kstride, nstride,
                                          ldc, sA, sW, sC, relu);
}

extern "C" void kernel_launch(void* const* d_in, const int* in_sizes, int n_in,
                              void* d_out, int out_size, void* d_ws, size_t ws_size,
                              hipStream_t stream) {
  const float* box_features = (const float*)d_in[0];
  const float* box_coords   = (const float*)d_in[1];
  const int*   box_labels   = (const int*)d_in[2];
  const float* box_scores   = (const float*)d_in[3];
  const float* W1  = (const float*)d_in[4];
  const float* b1  = (const float*)d_in[5];
  const float* W2  = (const float*)d_in[6];
  const float* b2  = (const float*)d_in[7];
  const float* W3  = (const float*)d_in[8];
  const float* b3  = (const float*)d_in[9];
  const float* Ws  = (const float*)d_in[10];
  const float* bs  = (const float*)d_in[11];
  const float* Wo  = (const float*)d_in[12];
  const float* bo  = (const float*)d_in[13];
  const float* Wsu = (const float*)d_in[14];
  const float* Wou = (const float*)d_in[15];
  float* out = (float*)d_out;
  float* ws  = (float*)d_ws;

  // workspace layout (floats)
  const long long ENC_SZ = (long long)B_ * N_ * R_;        // 2,097,152
  float* enc  = ws;
  float* U    = enc  + ENC_SZ;
  float* V    = U    + ENC_SZ;
  float* T    = V    + ENC_SZ;
  float* S    = T    + ENC_SZ;                              // B*16*1024
  float* CATH = S    + (long long)B_ * NH_ * REP_;          // B*16*2048
  float* CATO = CATH + (long long)B_ * NH_ * 2 * R_;        // B*64*2048
  float* Abuf = CATO + (long long)B_ * N_ * 2 * R_;         // B*16*64
  float* AT   = Abuf + (long long)B_ * NH_ * N_;            // B*64*16
  // transposed weights (WT[n][k])
  float* W1Tt = AT   + (long long)B_ * N_ * NH_;            // [1024][1024]
  float* W1Tb = W1Tt + (long long)REP_ * R_;
  float* WoT  = W1Tb + (long long)REP_ * R_;
  float* W2T  = WoT  + (long long)REP_ * R_;                // [512][1024]
  float* WsT  = W2T  + (long long)512 * REP_;
  float* WsuT = WsT  + (long long)REP_ * R_;                // [1024][2048]
  float* WouT = WsuT + (long long)R_ * 2 * R_;

  // pre-transpose all weights (call-invariant; ~34 MB, one HBM pass)
  transpose_kernel<<<2048, 256, 0, stream>>>(W1Tt, W1, R_, REP_);
  transpose_kernel<<<2048, 256, 0, stream>>>(W1Tb, W1 + (long long)R_ * REP_, R_, REP_);
  transpose_kernel<<<2048, 256, 0, stream>>>(WoT, Wo, R_, REP_);
  transpose_kernel<<<2048, 256, 0, stream>>>(W2T, W2, REP_, 512);
  transpose_kernel<<<2048, 256, 0, stream>>>(WsT, Ws, R_, REP_);
  transpose_kernel<<<2048, 256, 0, stream>>>(WsuT, Wsu, 2 * R_, R_);
  transpose_kernel<<<2048, 256, 0, stream>>>(WouT, Wou, 2 * R_, R_);

  // enc = box_features
  copy_rows_kernel<<<1024, 256, 0, stream>>>(enc, box_features, N_, R_,
                                             (long long)N_ * R_, R_,
                                             (long long)N_ * R_, R_, B_);

  for (int it = 0; it < 2; ++it) {
    // U = enc @ W1[:R], V = enc @ W1[R:]
    gemm(stream, enc, W1Tt, nullptr, U, N_, R_, REP_, R_, 1, R_, REP_,
         (long long)N_ * R_, 0, (long long)N_ * REP_, 0);
    gemm(stream, enc, W1Tb, nullptr, V, N_, R_, REP_, R_, 1, R_, REP_,
         (long long)N_ * R_, 0, (long long)N_ * REP_, 0);
    // T = relu(enc @ Wo + bo)
    gemm(stream, enc, WoT, bo, T, N_, R_, REP_, R_, 1, R_, REP_,
         (long long)N_ * R_, 0, (long long)N_ * REP_, 1);

    // A / A^T (diagonal entries stay 0)
    zero_kernel<<<64, 256, 0, stream>>>(Abuf, (long long)2 * B_ * NH_ * N_);
    pair_mlp_kernel<<<dim3(63, B_), dim3(128), (16 * HPAD + 16) * sizeof(float),
                      stream>>>(U, V, b1, W2T, b2, W3, b3, Abuf, AT);

    // CATH = [enc[:16], msg_h = A @ T]   (T row-major as K x N, K=64)
    copy_rows_kernel<<<256, 256, 0, stream>>>(CATH, enc, NH_, R_,
                                              (long long)NH_ * 2 * R_, 2 * R_,
                                              (long long)N_ * R_, R_, B_);
    gemm(stream, Abuf, T, nullptr, CATH + R_, NH_, N_, REP_, N_, REP_, 1, 2 * R_,
         (long long)NH_ * N_, (long long)N_ * REP_, (long long)NH_ * 2 * R_, 0);
    // enc[:16] = CATH @ Wsu
    gemm(stream, CATH, WsuT, nullptr, enc, NH_, 2 * R_, R_, 2 * R_, 1, 2 * R_, R_,
         (long long)NH_ * 2 * R_, 0, (long long)N_ * R_, 0);
    // S = relu(enc[:16] @ Ws + bs)
    gemm(stream, enc, WsT, bs, S, NH_, R_, REP_, R_, 1, R_, REP_,
         (long long)N_ * R_, 0, (long long)NH_ * REP_, 1);
    // CATO = [enc, msg_o = A^T @ S]   (S row-major as K x N, K=16)
    copy_rows_kernel<<<1024, 256, 0, stream>>>(CATO, enc, N_, R_,
                                               (long long)N_ * 2 * R_, 2 * R_,
                                               (long long)N_ * R_, R_, B_);
    gemm(stream, AT, S, nullptr, CATO + R_, N_, NH_, REP_, NH_, REP_, 1, 2 * R_,
         (long long)N_ * NH_, (long long)NH_ * REP_, (long long)N_ * 2 * R_, 0);
    // enc = CATO @ Wou
    gemm(stream, CATO, WouT, nullptr, enc, N_, 2 * R_, R_, 2 * R_, 1, 2 * R_, R_,
         (long long)N_ * 2 * R_, 0, (long long)N_ * R_, 0);
  }

  // outputs: pair gather + coords + prior
  finalize_kernel<<<B_ * M_, 128, 0, stream>>>(enc, box_coords, box_labels,
                                               box_scores, Abuf, out);
}